// MultiAdapterLinear_47356309406332
// MI455X (gfx1250) — compile-verified
//
#include <hip/hip_runtime.h>

#define D_IN   1024
#define D_OUT  1024
#define RANK   16
#define NTASK  32
#define BM     64
#define BN     64
#define KT     32
#define KTOT   (D_IN + NTASK * RANK)   /* 1536: base K + masked LoRA K */
#define NTILES (KTOT / KT)             /* 48 K-tiles total */
#define LDS_S  36                      /* padded row stride (floats): 16B-aligned, bank-conflict-free */

typedef float v2f __attribute__((ext_vector_type(2)));
typedef float v4f __attribute__((ext_vector_type(4)));
typedef float v8f __attribute__((ext_vector_type(8)));
typedef unsigned int u32x4 __attribute__((ext_vector_type(4)));
typedef int i32x4 __attribute__((ext_vector_type(4)));
typedef int i32x8 __attribute__((ext_vector_type(8)));

#if defined(__has_builtin)
#if __has_builtin(__builtin_amdgcn_tensor_load_to_lds) && __has_builtin(__builtin_amdgcn_s_wait_tensorcnt)
#define USE_TDM 1
#endif
#endif
#ifndef USE_TDM
#define USE_TDM 0
#endif

// ---------------------------------------------------------------------------
// Kernel 1: per-row LoRA down-projection  ax[b, r] = SCALING * (A[t_b] @ x[b])[r]
// One wave per row; A (2 MB) is L2-resident.
// ---------------------------------------------------------------------------
__global__ __launch_bounds__(256) void lora_down_kernel(
    const float* __restrict__ x, const int* __restrict__ task_ids,
    const float* __restrict__ A, float* __restrict__ ax)
{
  const int wave = threadIdx.x >> 5;
  const int lane = threadIdx.x & 31;
  const int row  = blockIdx.x * 8 + wave;
  const int t    = task_ids[row];

  const v4f* x4 = (const v4f*)(x + (size_t)row * D_IN);
  const v4f* A4 = (const v4f*)(A + (size_t)t * RANK * D_IN);

  float acc[RANK];
#pragma unroll
  for (int r = 0; r < RANK; ++r) acc[r] = 0.f;

#pragma unroll
  for (int it = 0; it < D_IN / 128; ++it) {
    v4f xv = x4[lane + 32 * it];
#pragma unroll
    for (int r = 0; r < RANK; ++r) {
      v4f av = A4[r * (D_IN / 4) + lane + 32 * it];
      acc[r] += xv.x * av.x + xv.y * av.y + xv.z * av.z + xv.w * av.w;
    }
  }

  float res = 0.f;
#pragma unroll
  for (int r = 0; r < RANK; ++r) {
    float v = acc[r];
#pragma unroll
    for (int off = 16; off > 0; off >>= 1) v += __shfl_xor(v, off, 32);
    if (lane == r) res = v;
  }
  if (lane < RANK)
    ax[(size_t)row * RANK + lane] = res * 2.0f;   // SCALING = 32/16
}

#if USE_TDM
// Issue a TDM 2D tile load: tile_k x tile_rows (f32) from global into LDS,
// padding each 32-DWORD row by 4 DWORDs -> LDS row stride = 36 floats.
// D# descriptor per CDNA5 ISA sec. 8.3/8.4 (group0 128b, group1 256b).
// This toolchain exposes the 6-arg builtin:
//   (u32x4 g0, i32x8 g1, i32x4 g2, i32x4 g3, i32x8 g4, i32 cpol)
__device__ __forceinline__ void tdm_load_tile(unsigned lds_byte,
                                              const float* gptr,
                                              unsigned tensor_dim0,
                                              unsigned tensor_dim1,
                                              unsigned tile_k,
                                              unsigned tile_rows,
                                              unsigned long long row_stride)
{
  unsigned long long ga = (unsigned long long)(uintptr_t)gptr;
  u32x4 g0;
  g0[0] = 1u;                                          // count=1, user D#
  g0[1] = lds_byte;                                    // lds_addr [63:32]
  g0[2] = (unsigned)(ga & 0xFFFFFFFFu);                // global_addr [95:64]
  g0[3] = (unsigned)((ga >> 32) & 0x01FFFFFFu)         // global_addr [120:96]
          | 0x80000000u;                               // type=2 ("image")
  i32x8 g1;
  g1[0] = (2 << 16)      // data_size = 4 bytes
        | (1 << 20)      // pad_enable
        | (4 << 22)      // pad_interval: 2^(4+1)=32 DWORDs
        | (3 << 25);     // pad_amount:  3+1 = 4 DWORDs  -> stride 36
  g1[1] = (int)((tensor_dim0 & 0xFFFFu) << 16);                        // dim0 lo
  g1[2] = (int)((tensor_dim0 >> 16) & 0xFFFFu)
        | (int)((tensor_dim1 & 0xFFFFu) << 16);                        // dim0 hi | dim1 lo
  g1[3] = (int)((tensor_dim1 >> 16) & 0xFFFFu)
        | (int)((tile_k & 0xFFFFu) << 16);                             // dim1 hi | tile_dim0
  g1[4] = (int)(tile_rows & 0xFFFFu);                                  // tile_dim1 | tile_dim2=0
  g1[5] = (int)(row_stride & 0xFFFFFFFFull);                           // dim0_stride lo32
  g1[6] = (int)((row_stride >> 32) & 0xFFFFull);                       // dim0_stride hi16
  g1[7] = 0;
  i32x4 g2 = {0, 0, 0, 0};
  i32x4 g3 = {0, 0, 0, 0};
  i32x8 g4 = {0, 0, 0, 0, 0, 0, 0, 0};
  __builtin_amdgcn_tensor_load_to_lds(g0, g1, g2, g3, g4, 0);
}
#endif

// ---------------------------------------------------------------------------
// Kernel 2: fused GEMM  out = [x | axz] @ [W^T ; Bflat] + bias  via
// V_WMMA_F32_16X16X4_F32 with double-buffered LDS and async TDM staging
// overlapped with compute.  K = 1024 (base) + 512 (masked LoRA contraction).
// 64x64 tile / 256-thread WG, 8 waves in 4(M) x 2(N), 16x32 C tile per wave.
// ---------------------------------------------------------------------------
__global__ __launch_bounds__(256) void fused_gemm_kernel(
    const float* __restrict__ x, const int* __restrict__ task_ids,
    const float* __restrict__ W, const float* __restrict__ bias,
    const float* __restrict__ Bup, const float* __restrict__ ax,
    float* __restrict__ out)
{
  __shared__ float Xs[2][BM * LDS_S];
  __shared__ float Ws[2][BN * LDS_S];

  const int b0   = blockIdx.y * BM;
  const int o0   = blockIdx.x * BN;
  const int tid  = threadIdx.x;
  const int lane = tid & 31;
  const int wv   = tid >> 5;
  const int wm   = (wv & 3) * 16;   // wave M offset in tile
  const int wn   = (wv >> 2) * 32;  // wave N offset in tile

  v8f c0 = {};
  v8f c1 = {};

  // Stage K-tile `it` into buffers Xd/Wd.  Base region uses the TDM (async,
  // issued by wave 0 only; EXEC-independent DMA); LoRA region is manual.
  auto stage = [&](int it, float* Xd, float* Wd) {
    const int k0 = it * KT;
    if (k0 < D_IN) {
#if USE_TDM
      if (wv == 0) {
        tdm_load_tile((unsigned)(uintptr_t)Xd,
                      &x[(size_t)b0 * D_IN + k0], D_IN - k0, BM, KT, BM, D_IN);
        tdm_load_tile((unsigned)(uintptr_t)Wd,
                      &W[(size_t)o0 * D_IN + k0], D_IN - k0, BN, KT, BN, D_IN);
      }
#else
#pragma unroll
      for (int i = 0; i < (BM * KT) / (256 * 4); ++i) {   // b128 staging
        int e = tid + 256 * i;
        int m = e >> 3, q = e & 7;                         // 8 float4 per row
        *(v4f*)&Xd[m * LDS_S + q * 4] =
            *(const v4f*)&x[(size_t)(b0 + m) * D_IN + k0 + q * 4];
        *(v4f*)&Wd[m * LDS_S + q * 4] =
            *(const v4f*)&W[(size_t)(o0 + m) * D_IN + k0 + q * 4];
      }
      if (k0 + KT < D_IN) {   // hint next x tile (global_prefetch_b8)
        int m = tid >> 5, kk = tid & 31;
        __builtin_prefetch(&x[(size_t)(b0 + m) * D_IN + k0 + KT + kk], 0, 1);
      }
#endif
    } else {
      // LoRA region: axz[b,(t,r)] masked by task id; Bflat[(t,r),o] = B[t][o][r]
      const int kbase = k0 - D_IN;
#pragma unroll
      for (int i = 0; i < (BM * KT) / (256 * 4); ++i) {
        int e  = tid + 256 * i;
        int m  = e >> 3, q = e & 7;
        int kk = q * 4;                 // 4 consecutive r within one task block
        int k2 = kbase + kk;
        int tt = k2 >> 4, r = k2 & 15;
        int brow = b0 + m;
        // branchless mask: unconditional L2-resident loads + select
        v4f axv = *(const v4f*)&ax[(size_t)brow * RANK + r];
        int  tk = task_ids[brow];
        v4f zero = {};
        *(v4f*)&Xd[m * LDS_S + kk] = (tt == tk) ? axv : zero;
        *(v4f*)&Wd[m * LDS_S + kk] =
            *(const v4f*)&Bup[((size_t)tt * D_OUT + (o0 + m)) * RANK + r];
      }
    }
  };

  // ---- prologue: stage tile 0, wait, release ----
  stage(0, Xs[0], Ws[0]);
#if USE_TDM
  if (wv == 0) __builtin_amdgcn_s_wait_tensorcnt(0);
#endif
  __syncthreads();

  // ---- main pipeline: stage(it+1) overlapped with compute(it) ----
  const int koff = (lane >> 4) * 2;
  const int mrow = wm + (lane & 15);
  const int n0   = wn + (lane & 15);
  const int n1   = n0 + 16;

  for (int it = 0; it < NTILES; ++it) {
    const int p = it & 1;
    if (it + 1 < NTILES)
      stage(it + 1, Xs[p ^ 1], Ws[p ^ 1]);    // async TDM / manual into other buffer

    const float* Xp = Xs[p];
    const float* Wp = Ws[p];
    // WMMA over this K tile (16x16x4 f32)
    // A 16x4: lanes 0-15 hold K=0,1; lanes 16-31 hold K=2,3 (per M row)
    // B 4x16: lanes 0-15 hold K=0,1; lanes 16-31 hold K=2,3 (per N col)
#pragma unroll
    for (int kb = 0; kb < KT; kb += 4) {
      v2f a, bA, bB;
      a.x  = Xp[mrow * LDS_S + kb + koff];
      a.y  = Xp[mrow * LDS_S + kb + koff + 1];
      bA.x = Wp[n0 * LDS_S + kb + koff];
      bA.y = Wp[n0 * LDS_S + kb + koff + 1];
      bB.x = Wp[n1 * LDS_S + kb + koff];
      bB.y = Wp[n1 * LDS_S + kb + koff + 1];
      c0 = __builtin_amdgcn_wmma_f32_16x16x4_f32(false, a, false, bA,
                                                 (short)0, c0, false, false);
      c1 = __builtin_amdgcn_wmma_f32_16x16x4_f32(false, a, false, bB,
                                                 (short)0, c1, false, false);
    }

    __syncthreads();   // all waves done reading buf p; manual stores to buf p^1 done
#if USE_TDM
    if (wv == 0) __builtin_amdgcn_s_wait_tensorcnt(0);   // TDM for tile it+1 done
    __syncthreads();                                     // release next tile
#endif
  }

  // ---- epilogue: C layout = VGPR v: lanes 0-15 -> M=v, lanes 16-31 -> M=v+8
  const int nlo   = lane & 15;
  const int mhalf = (lane >> 4) * 8;
  const float bias0 = bias[o0 + wn + nlo];
  const float bias1 = bias[o0 + wn + 16 + nlo];
#pragma unroll
  for (int v = 0; v < 8; ++v) {
    int m = b0 + wm + mhalf + v;
    out[(size_t)m * D_OUT + (o0 + wn + nlo)]      = c0[v] + bias0;
    out[(size_t)m * D_OUT + (o0 + wn + 16 + nlo)] = c1[v] + bias1;
  }
}

// ---------------------------------------------------------------------------
extern "C" void kernel_launch(void* const* d_in, const int* in_sizes, int n_in,
                              void* d_out, int out_size, void* d_ws, size_t ws_size,
                              hipStream_t stream) {
  const float* x        = (const float*)d_in[0];
  const int*   task_ids = (const int*)d_in[1];
  const float* W        = (const float*)d_in[2];
  const float* bias     = (const float*)d_in[3];
  const float* A        = (const float*)d_in[4];
  const float* Bup      = (const float*)d_in[5];
  float*       out      = (float*)d_out;
  float*       ax       = (float*)d_ws;          // [B, 16] f32 scratch (2 MB)

  const int Brows = in_sizes[1];                 // 32768 rows

  lora_down_kernel<<<Brows / 8, 256, 0, stream>>>(x, task_ids, A, ax);

  dim3 grid(D_OUT / BN, Brows / BM);             // (16, 512)
  fused_gemm_kernel<<<grid, 256, 0, stream>>>(x, task_ids, W, bias, Bup, ax, out);
}